// P1_58591943852133
// MI455X (gfx1250) — compile-verified
//
#include <hip/hip_runtime.h>
#include <stdint.h>

// MI455X / gfx1250 streaming MLP (1 -> 2 -> [2x2]*8 -> 1, ReLU).
// Memory-bound: 8 B/row * 8M rows = 64 MB @ 23.3 TB/s ~ 2.8 us.
// 2x2 matrices make WMMA a >4x pessimization (<=12.5% MAC utilization), so
// compute runs on V_PK_FMA_F32 (VOP3P) in an element-pair, feature-major
// layout. Weight splats are forced into VGPR pairs at their definition
// (empty asm "+v" breaks uniformity -> no SGPR spills); bias splats stay in
// SGPR pairs (the one legal scalar source per VOP3P). Packed-f32 VOP3P
// requires OPSEL_HI == 0 on this target, so the asm spells op_sel/op_sel_hi
// as explicit zeros. Input staging uses the CDNA5 async-to-LDS DMA path
// (ASYNCcnt) with a 4-slot ring buffer, 3 tiles of prefetch in flight.

typedef float v2f __attribute__((ext_vector_type(2)));
typedef float v4f __attribute__((ext_vector_type(4)));

#define NBLK 2048
#define NTHR 256

// Force a (uniform) value to be defined in a VGPR pair. Emits no code by
// itself; materializes one s->v copy at setup and kills the SGPR live range.
__device__ __forceinline__ v2f to_vgpr(v2f a) {
  asm("" : "+v"(a));
  return a;
}

// d = wv * hv + bs, packed f32; weight + h in VGPR pairs, bias from an SGPR
// pair. Packed-FP32 ops mandate OPSEL_HI == 0 (parser default is rejected).
__device__ __forceinline__ v2f pk_fma_vvs(v2f wv, v2f hv, v2f bs) {
  v2f d;
  asm("v_pk_fma_f32 %0, %1, %2, %3 op_sel:[0,0,0] op_sel_hi:[0,0,0]"
      : "=v"(d) : "v"(wv), "v"(hv), "s"(bs));
  return d;
}
// d = wv * hv + cv, all VGPR pairs (accumulator chain).
__device__ __forceinline__ v2f pk_fma_vvv(v2f wv, v2f hv, v2f cv) {
  v2f d;
  asm("v_pk_fma_f32 %0, %1, %2, %3 op_sel:[0,0,0] op_sel_hi:[0,0,0]"
      : "=v"(d) : "v"(wv), "v"(hv), "v"(cv));
  return d;
}

__device__ __forceinline__ v2f relu2(v2f a) {
  a.x = __builtin_fmaxf(a.x, 0.f);
  a.y = __builtin_fmaxf(a.y, 0.f);
  return a;
}

struct Wts {
  // Weights pre-splatted (w,w), resident in VGPR pairs.
  v2f wi0, wi1;                       // input layer weights
  v2f w00[8], w01[8], w10[8], w11[8]; // hidden weights W[k][i][j]
  v2f wo0, wo1;                       // output weights
  // Biases pre-splatted (b,b), resident in SGPR pairs.
  v2f bi0, bi1, b0[8], b1[8], bo;
};

// Two rows at once, feature-major: X = (x_e0, x_e1) is a natural VGPR pair.
// Returns (out_e0, out_e1) in a VGPR pair.
__device__ __forceinline__ v2f mlp_pair(v2f X, const Wts& W) {
  v2f H0 = relu2(pk_fma_vvs(W.wi0, X, W.bi0));
  v2f H1 = relu2(pk_fma_vvs(W.wi1, X, W.bi1));
#pragma unroll
  for (int k = 0; k < 8; ++k) {
    // y[i] = W[k][i][0]*h0 + W[k][i][1]*h1 + b[k][i]   (torch h @ W.T + b)
    v2f T0 = pk_fma_vvs(W.w01[k], H1, W.b0[k]);
    T0 = pk_fma_vvv(W.w00[k], H0, T0);
    v2f T1 = pk_fma_vvs(W.w11[k], H1, W.b1[k]);
    T1 = pk_fma_vvv(W.w10[k], H0, T1);
    H0 = relu2(T0);
    H1 = relu2(T1);
  }
  v2f O = pk_fma_vvs(W.wo1, H1, W.bo);
  O = pk_fma_vvv(W.wo0, H0, O);
  return relu2(O);
}

__device__ __forceinline__ v2f splat2(float v) { return (v2f){v, v}; }

// Async DMA: each active lane copies 16 B from global into its LDS slot.
// Tracked by ASYNCcnt; completes in order.
__device__ __forceinline__ void async_ld16(unsigned lds_off, const float* g) {
  unsigned long long ga = (unsigned long long)(uintptr_t)g;
  asm volatile("global_load_async_to_lds_b128 %0, %1, off"
               :: "v"(lds_off), "v"(ga)
               : "memory");
}

__global__ __launch_bounds__(NTHR) void P1_mlp_stream(
    const float* __restrict__ x, const float* __restrict__ w_in,
    const float* __restrict__ b_in, const float* __restrict__ w_hid,
    const float* __restrict__ b_hid, const float* __restrict__ w_out,
    const float* __restrict__ b_out, float* __restrict__ out, int n) {
  __shared__ __align__(16) float stage[4][NTHR][4];  // 16 KB, 4-slot ring

  // Uniform scalar loads -> SGPRs; weight splats immediately moved to VGPRs
  // (to_vgpr) so their scalar live ranges die at setup. Biases stay scalar.
  Wts W;
  W.wi0 = to_vgpr(splat2(w_in[0]));  W.wi1 = to_vgpr(splat2(w_in[1]));
  W.bi0 = splat2(b_in[0]);           W.bi1 = splat2(b_in[1]);
#pragma unroll
  for (int k = 0; k < 8; ++k) {
    W.w00[k] = to_vgpr(splat2(w_hid[4 * k + 0]));
    W.w01[k] = to_vgpr(splat2(w_hid[4 * k + 1]));
    W.w10[k] = to_vgpr(splat2(w_hid[4 * k + 2]));
    W.w11[k] = to_vgpr(splat2(w_hid[4 * k + 3]));
    W.b0[k] = splat2(b_hid[2 * k + 0]);
    W.b1[k] = splat2(b_hid[2 * k + 1]);
  }
  W.wo0 = to_vgpr(splat2(w_out[0])); W.wo1 = to_vgpr(splat2(w_out[1]));
  W.bo  = splat2(b_out[0]);

  const int nvec = n >> 2;  // float4 tiles
  if (nvec > 0) {
    const int nvecm1 = nvec - 1;
    const int T = gridDim.x * blockDim.x;
    const int i0 = blockIdx.x * blockDim.x + threadIdx.x;

    // Low 32 bits of a generic LDS pointer == raw LDS byte offset, which is
    // what the async instruction's LDS-address VGPR expects.
    const unsigned base = (unsigned)(uintptr_t)&stage[0][0][0];
    const unsigned lane_off = threadIdx.x * 16u;

    // Prologue: fill 3 ring slots. Out-of-range lanes clamp to the last
    // valid tile (harmless; keeps per-wave outstanding counts uniform).
#pragma unroll
    for (int c = 0; c < 3; ++c) {
      int idx = i0 + c * T;
      idx = idx < nvecm1 ? idx : nvecm1;
      async_ld16(base + (unsigned)c * (NTHR * 16u) + lane_off,
                 x + 4 * (size_t)idx);
    }

    int k = 0;
    for (int t = i0; t < nvec; t += T, ++k) {
      // Issue prefetch 3 tiles ahead into the slot consumed last iteration
      // (its ds_load already completed before this point in program order).
      int pf = t + 3 * T;
      pf = pf < nvecm1 ? pf : nvecm1;
      async_ld16(base + (unsigned)((k + 3) & 3) * (NTHR * 16u) + lane_off,
                 x + 4 * (size_t)pf);
      // 4 outstanding after issue; <=3 retires exactly the tile we consume.
      asm volatile("s_wait_asynccnt 0x3" ::: "memory");

      const v4f xv = *(const v4f*)&stage[k & 3][threadIdx.x][0];  // ds_load_b128
      const v2f o01 = mlp_pair((v2f){xv.x, xv.y}, W);
      const v2f o23 = mlp_pair((v2f){xv.z, xv.w}, W);
      const v4f ov = (v4f){o01.x, o01.y, o23.x, o23.y};
      // Single-pass stream: non-temporal store, don't pollute L2.
      __builtin_nontemporal_store(ov, (v4f*)(out + 4 * (size_t)t));
    }
  }

  // Tail (n % 4) — N=8M is divisible by 4, but stay generic.
  int rem = n & 3;
  if (blockIdx.x == 0 && (int)threadIdx.x < rem) {
    int idx = (n & ~3) + (int)threadIdx.x;
    float xs = x[idx];
    out[idx] = mlp_pair((v2f){xs, xs}, W).x;
  }
}

extern "C" void kernel_launch(void* const* d_in, const int* in_sizes, int n_in,
                              void* d_out, int out_size, void* d_ws, size_t ws_size,
                              hipStream_t stream) {
  const float* x     = (const float*)d_in[0];
  const float* w_in  = (const float*)d_in[1];
  const float* b_in  = (const float*)d_in[2];
  const float* w_hid = (const float*)d_in[3];
  const float* b_hid = (const float*)d_in[4];
  const float* w_out = (const float*)d_in[5];
  const float* b_out = (const float*)d_in[6];
  float* out = (float*)d_out;
  int n = in_sizes[0];

  P1_mlp_stream<<<NBLK, NTHR, 0, stream>>>(x, w_in, b_in, w_hid, b_hid,
                                           w_out, b_out, out, n);
  (void)n_in; (void)out_size; (void)d_ws; (void)ws_size;
}